// Cif_19421842112898
// MI455X (gfx1250) — compile-verified
//
#include <hip/hip_runtime.h>
#include <stdint.h>

// Problem constants (fixed by setup_inputs): B=32, S=2048, C=512, Tmax=512
constexpr int BB   = 32;
constexpr int SS   = 2048;
constexpr int CH   = 512;
constexpr int TMX  = 512;
constexpr int SP   = SS + 16;          // padded per-batch stride for ws tables
constexpr float THETA = 1.0f;          // FIRING_THRESHOLD
constexpr float EPSF  = 1e-4f;         // EPS
constexpr float SENT  = 4096.0f;       // sentinel csum beyond S (clamps to Tmax)

typedef __attribute__((ext_vector_type(2))) float v2f;
typedef __attribute__((ext_vector_type(8))) float v8f;

// ---------------------------------------------------------------------------
// Kernel 1: per-batch prep.
//  - w_sum, scaled csum (zero-prepended + sentinel-padded, to ws)
//  - per-source coefficient metadata float4{li, nf, rw, lw} (to ws)
//  - right_w / left_w outputs, durations run-length pass
// One 1024-thread block (32 waves) per batch row; each thread owns s=2t,2t+1.
// ---------------------------------------------------------------------------
__global__ __launch_bounds__(1024)
void cif_prep(const float* __restrict__ in_w,
              const unsigned char* __restrict__ pad,
              const int* __restrict__ tlen,
              float* __restrict__ csum_ws,      // per batch: [0]=0, [1..S]=csum, pad=SENT
              float4* __restrict__ meta_ws,     // per batch: [0..S-1]=meta, pad rows
              float* __restrict__ out_wsum,
              float* __restrict__ out_rw,
              float* __restrict__ out_lw,
              float* __restrict__ out_dur)
{
    const int b   = blockIdx.x;
    const int tid = threadIdx.x;          // 0..1023
    const int s0  = tid * 2;
    const int s1  = s0 + 1;

    __shared__ float sf[1024];
    __shared__ int   si[1024];
    __shared__ int   sseq[SS + 1];
    __shared__ int   counts[TMX + 1];
    __shared__ int   s_alen;
    __shared__ int   s_dursum;

    const float*         wrow = in_w + (size_t)b * SS;
    const unsigned char* prow = pad  + (size_t)b * SS;
    float*  csE   = csum_ws + (size_t)b * SP;
    float4* metab = meta_ws + (size_t)b * SP;

    float w0 = fminf(fmaxf(wrow[s0], 0.f), 1.f);
    float w1 = fminf(fmaxf(wrow[s1], 0.f), 1.f);
    const bool p0 = prow[s0] != 0;
    const bool p1 = prow[s1] != 0;
    if (p0) w0 = 0.f;
    if (p1) w1 = 0.f;

    if (tid == 0) { s_alen = 0; s_dursum = 0; }
    __syncthreads();
    atomicAdd(&s_alen, (p0 ? 0 : 1) + (p1 ? 0 : 1));

    // ---- inclusive scan of per-thread pair sums (Hillis-Steele in LDS) ----
    const float pair = w0 + w1;
    sf[tid] = pair;
    __syncthreads();
    #pragma unroll
    for (int off = 1; off < 1024; off <<= 1) {
        float v = (tid >= off) ? sf[tid - off] : 0.f;
        __syncthreads();
        sf[tid] += v;
        __syncthreads();
    }
    const float incl  = sf[tid];
    const float wsum  = sf[1023];
    const int   tlI   = tlen[b];
    const float scale = (THETA * (float)tlI + EPSF) / wsum;

    const float exclr = incl - pair;               // raw exclusive prefix
    const float cprev = exclr * scale;             // csum[s0-1] (0 at tid==0)
    const float c0    = (exclr + w0) * scale;      // csum[s0]
    const float c1    = (exclr + w0 + w1) * scale; // csum[s1]

    // zero-prepended, sentinel-padded csum table
    csE[s0 + 1] = c0;
    csE[s1 + 1] = c1;
    if (tid == 0) { csE[0] = 0.f; out_wsum[b] = wsum; }
    if (tid < 15) csE[SS + 1 + tid] = SENT;
    if (tid < 16) metab[SS + tid] = make_float4((float)TMX, 0.f, 0.f, 0.f);

    // ---- per-source meta + right_w / left_w outputs (theta = 1) ----
    {
        const float cl = cprev, cr = c0;
        int ri = min(max((int)floorf(cr), 0), TMX);
        int li = min(max((int)floorf(cl), 0), TMX);
        int nf = ri - li;
        int ex = max(nf - 1, 0);
        float rw = (nf > 0) ? (cr - (float)ri * THETA) : 0.f;
        float lw = (cr - cl) - rw - (float)ex * THETA;
        out_rw[(size_t)b * SS + s0] = rw;
        out_lw[(size_t)b * SS + s0] = lw;
        metab[s0] = make_float4((float)li, (float)nf, rw, lw);
    }
    {
        const float cl = c0, cr = c1;
        int ri = min(max((int)floorf(cr), 0), TMX);
        int li = min(max((int)floorf(cl), 0), TMX);
        int nf = ri - li;
        int ex = max(nf - 1, 0);
        float rw = (nf > 0) ? (cr - (float)ri * THETA) : 0.f;
        float lw = (cr - cl) - rw - (float)ex * THETA;
        out_rw[(size_t)b * SS + s1] = rw;
        out_lw[(size_t)b * SS + s1] = lw;
        metab[s1] = make_float4((float)li, (float)nf, rw, lw);
    }

    // ---- durations: seq = min(round(csum), tl) ++ [tl]; run-length encode ----
    sseq[s0] = min((int)rintf(c0), tlI);
    sseq[s1] = min((int)rintf(c1), tlI);
    if (tid == 1023) sseq[SS] = tlI;
    __syncthreads();

    const int ca = (sseq[s1]     != sseq[s0]) ? 1 : 0;
    const int cb = (sseq[s1 + 1] != sseq[s1]) ? 1 : 0;
    si[tid] = ca + cb;
    __syncthreads();
    #pragma unroll
    for (int off = 1; off < 1024; off <<= 1) {
        int v = (tid >= off) ? si[tid - off] : 0;
        __syncthreads();
        si[tid] += v;
        __syncthreads();
    }
    const int inclI = si[tid];
    const int m     = si[1023];        // run_id[S] = number of changes
    const int rid_b = inclI - cb;      // run_id at i = 2t+1
    const int rid_c = inclI;           // run_id at i = 2t+2

    if (tid <= TMX) counts[tid] = 0;
    __syncthreads();
    if (tid == 0) atomicAdd(&counts[0], 1);   // run_id[0] == 0
    atomicAdd(&counts[rid_b], 1);
    atomicAdd(&counts[rid_c], 1);
    __syncthreads();

    if (tid <= TMX) {
        int d = (tid < m) ? counts[tid] : 0;
        counts[tid] = d;
        atomicAdd(&s_dursum, d);
    }
    __syncthreads();
    if (tid == 0) {
        const int L = (m >= tlI) ? m : m + 1;   // L-1 <= 511 here
        counts[L - 1] += s_alen - s_dursum;
    }
    __syncthreads();
    if (tid < TMX) out_dur[(size_t)b * TMX + tid] = (float)counts[tid];
}

// ---------------------------------------------------------------------------
// Branch-free CIF coefficient from precomputed meta {li, nf, rw, lw}.
// Sentinel rows (li=Tmax, nf=0, lw=0) yield 0 for all t < Tmax.
// ---------------------------------------------------------------------------
__device__ __forceinline__ float cif_coef(const float4 m, const float tf)
{
    float c = (tf == m.x) ? m.w : 0.f;                       // left_w at li
    c += (m.y > 0.f && tf == m.x + m.y) ? m.z : 0.f;         // right_w at ri=li+nf
    c += (m.y >= 2.f && tf == m.x + 1.f) ? 1.0f : 0.f;       // extra k=1 (theta)
    c += (m.y >= 3.f && tf == m.x + 2.f) ? 1.0f : 0.f;       // extra k=2
    return c;
}

// ---------------------------------------------------------------------------
// Kernel 2: banded-matmul scatter via V_WMMA_F32_16X16X4_F32.
// grid = (Tmax/16, C/128, B); 8 waves/block; each wave owns a 16x16 (t,c) tile.
// output tile = A(16 x K) * inputs(K x 16), K-chunks of 4, unrolled x2.
// ---------------------------------------------------------------------------
__global__ __launch_bounds__(256)
void cif_wmma(const float* __restrict__ inputs,
              const float* __restrict__ csum_ws,
              const float4* __restrict__ meta_ws,
              float* __restrict__ out,
              float* __restrict__ out_delay)
{
    const int b     = blockIdx.z;
    const int t0    = blockIdx.x * 16;
    const int wave  = threadIdx.x >> 5;
    const int lane  = threadIdx.x & 31;
    const int cbase = blockIdx.y * 128 + wave * 16;

    const float*  cs1   = csum_ws + (size_t)b * SP + 1;   // cs1[s] = csum[s]
    const float4* metab = meta_ws + (size_t)b * SP;
    const float*  inb   = inputs  + (size_t)b * SS * CH;

    // first s with csum[s] >= t0  (right_idx[s] >= t0)
    int lo = 0, hi = SS;
    const float thr0 = (float)t0;
    while (lo < hi) { int mid = (lo + hi) >> 1; if (cs1[mid] < thr0) lo = mid + 1; else hi = mid; }
    const int s_lo = lo;
    // first s with csum[s-1] >= t0+16  (left_idx[s] >= t0+16)
    lo = 0; hi = SS;
    const float thr1 = (float)(t0 + 16);
    while (lo < hi) { int mid = (lo + hi) >> 1; if (cs1[mid] < thr1) lo = mid + 1; else hi = mid; }
    const int s_hi = min(lo + 1, SS);

    v8f acc = {0.f, 0.f, 0.f, 0.f, 0.f, 0.f, 0.f, 0.f};
    float dly = 0.f;

    const int   mrow  = lane & 15;          // target row within tile (both halves)
    const int   klo   = (lane >> 4) << 1;   // K pair: lanes 0-15 -> {0,1}, 16-31 -> {2,3}
    const float tf    = (float)(t0 + mrow);

    float sf1 = (float)(s_lo + klo + 1);    // src_range of first owned source
    for (int sb = s_lo; sb < s_hi; sb += 8) {
        // ---- chunk 0: sources sb .. sb+3 ----
        {
            const int sA = sb + klo;
            const int sB = sA + 1;
            const float4 mA = metab[sA];
            const float4 mB = metab[sB];
            v2f a;  a.x = cif_coef(mA, tf);  a.y = cif_coef(mB, tf);
            const int sAc = min(sA, SS - 1);
            const int sBc = min(sB, SS - 1);
            v2f bm;
            bm.x = inb[(size_t)sAc * CH + cbase + mrow];
            bm.y = inb[(size_t)sBc * CH + cbase + mrow];
            acc = __builtin_amdgcn_wmma_f32_16x16x4_f32(
                false, a, false, bm, (short)0, acc, false, false);
            dly += a.x * sf1 + a.y * (sf1 + 1.f);
        }
        // ---- chunk 1: sources sb+4 .. sb+7 ----
        {
            const int sA = sb + 4 + klo;
            const int sB = sA + 1;
            const float4 mA = metab[sA];
            const float4 mB = metab[sB];
            v2f a;  a.x = cif_coef(mA, tf);  a.y = cif_coef(mB, tf);
            const int sAc = min(sA, SS - 1);
            const int sBc = min(sB, SS - 1);
            v2f bm;
            bm.x = inb[(size_t)sAc * CH + cbase + mrow];
            bm.y = inb[(size_t)sBc * CH + cbase + mrow];
            acc = __builtin_amdgcn_wmma_f32_16x16x4_f32(
                false, a, false, bm, (short)0, acc, false, false);
            dly += a.x * (sf1 + 4.f) + a.y * (sf1 + 5.f);
        }
        sf1 += 8.f;
    }

    // store 16x16 D tile: VGPR r -> M = r (+8 for upper lanes), N = lane&15
    float* outb = out + ((size_t)b * TMX + t0) * CH + cbase;
    const int mbase = (lane >> 4) * 8;
    #pragma unroll
    for (int r = 0; r < 8; ++r) {
        outb[(size_t)(mbase + r) * CH + mrow] = acc[r];
    }

    // delay: combine the two lane-halves (K pairs), one writer per (b, t)
    dly += __shfl_xor(dly, 16, 32);
    if (blockIdx.y == 0 && wave == 0 && lane < 16) {
        out_delay[(size_t)b * TMX + t0 + mrow] = dly;
    }
}

// ---------------------------------------------------------------------------
// Host launcher
// ---------------------------------------------------------------------------
extern "C" void kernel_launch(void* const* d_in, const int* in_sizes, int n_in,
                              void* d_out, int out_size, void* d_ws, size_t ws_size,
                              hipStream_t stream)
{
    const float*         inputs = (const float*)d_in[0];          // (B,S,C)
    const float*         in_w   = (const float*)d_in[1];          // (B,S)
    const unsigned char* pad    = (const unsigned char*)d_in[2];  // (B,S) bool
    const int*           tlen   = (const int*)d_in[3];            // (B,)
    // d_in[4] = Tmax scalar (512), baked into constants

    float* out       = (float*)d_out;                       // (B,Tmax,C)
    float* out_delay = out + (size_t)BB * TMX * CH;         // (B,Tmax)
    float* out_dur   = out_delay + (size_t)BB * TMX;        // (B,Tmax)
    float* out_wsum  = out_dur + (size_t)BB * TMX;          // (B,)
    float* out_rw    = out_wsum + BB;                       // (B,S)
    float* out_lw    = out_rw + (size_t)BB * SS;            // (B,S)

    float*  csum_ws = (float*)d_ws;                         // BB*SP floats (~264 KB)
    float4* meta_ws = (float4*)((char*)d_ws + (size_t)BB * SP * sizeof(float)); // ~1 MB

    cif_prep<<<dim3(BB), dim3(1024), 0, stream>>>(
        in_w, pad, tlen, csum_ws, meta_ws, out_wsum, out_rw, out_lw, out_dur);

    dim3 grid(TMX / 16, CH / 128, BB);
    cif_wmma<<<grid, dim3(256), 0, stream>>>(inputs, csum_ws, meta_ws, out, out_delay);
}